// GraphAttentionLayer_38474317038007
// MI455X (gfx1250) — compile-verified
//
#include <hip/hip_runtime.h>

#define N_NODES 8192
#define IN_FEAT 512
#define OUT_FEAT 128
#define ALPHA_SLOPE 0.2f
#define NEG_FILL (-9e15f)
#define CHUNK 128
#define PSTR 132   // padded LDS row stride: 528B/row -> 16B aligned, conflict-free A-frag reads

typedef __attribute__((ext_vector_type(2))) float v2f;
typedef __attribute__((ext_vector_type(8))) float v8f;

__device__ __forceinline__ unsigned fkey(float f) {
    unsigned u = __float_as_uint(f);
    return (u & 0x80000000u) ? ~u : (u | 0x80000000u);
}
__device__ __forceinline__ float unkey(unsigned k) {
    unsigned u = (k & 0x80000000u) ? (k & 0x7fffffffu) : ~k;
    return __uint_as_float(u);
}
__device__ __forceinline__ float leaky(float x) {
    return x >= 0.f ? x : ALPHA_SLOPE * x;
}
// low 32 bits of a generic shared-aperture address == LDS byte address
__device__ __forceinline__ unsigned lds_addr32(const void* p) {
    return (unsigned)(unsigned long long)p;
}
// async copy 16B global -> LDS (per-lane), tracked by ASYNCcnt
__device__ __forceinline__ void async_g2l_b128(unsigned lds, const float* gaddr) {
    asm volatile("global_load_async_to_lds_b128 %0, %1, off"
                 :
                 : "v"(lds), "v"(gaddr)
                 : "memory");
}

// ---------------- utility: zero a u32 region ----------------
__global__ void zero_kernel(unsigned* p, int n) {
    int i = blockIdx.x * blockDim.x + threadIdx.x;
    for (; i < n; i += gridDim.x * blockDim.x) p[i] = 0u;
}

// ---------------- radix-select histogram (3 passes: 12/10/10 bits) ----------------
__global__ void hist_kernel(const float* __restrict__ adj, unsigned* __restrict__ hist,
                            const unsigned* __restrict__ sel, int pass) {
    __shared__ unsigned lh[4096];
    int t = threadIdx.x;
    int nb = (pass == 0) ? 4096 : 1024;
    for (int i = t; i < nb; i += blockDim.x) lh[i] = 0u;
    __syncthreads();
    unsigned prefix = (pass > 0) ? sel[0] : 0u;
    const long long total = (long long)N_NODES * N_NODES;
    long long stride = (long long)gridDim.x * blockDim.x;
    for (long long idx = (long long)blockIdx.x * blockDim.x + t; idx < total; idx += stride) {
        if (idx + stride < total)
            __builtin_prefetch(adj + idx + stride, 0, 0);  // global_prefetch_b8
        unsigned key = fkey(adj[idx]);
        if (pass == 0) {
            atomicAdd(&lh[key >> 20], 1u);
        } else if (pass == 1) {
            if ((key >> 20) == prefix) atomicAdd(&lh[(key >> 10) & 1023u], 1u);
        } else {
            if ((key >> 10) == prefix) atomicAdd(&lh[key & 1023u], 1u);
        }
    }
    __syncthreads();
    for (int i = t; i < nb; i += blockDim.x)
        if (lh[i]) atomicAdd(&hist[i], lh[i]);
}

// scan histogram top-down, refine prefix/rank, zero histogram for next use
__global__ void scan_kernel(unsigned* hist, unsigned* sel, const int* kptr, int pass) {
    if (threadIdx.x != 0 || blockIdx.x != 0) return;
    int nb = (pass == 0) ? 4096 : 1024;
    unsigned long long rank;
    if (pass == 0) {
        long long kk = (long long)kptr[0] * (long long)N_NODES;
        long long tot = (long long)N_NODES * N_NODES;
        if (kk > tot) kk = tot;
        if (kk < 1) kk = 1;
        rank = (unsigned long long)kk;
    } else {
        rank = sel[1];
    }
    unsigned long long cum = 0;
    unsigned b = 0;
    for (int i = nb - 1; i >= 0; --i) {
        unsigned c = hist[i];
        if (cum + c >= rank) { b = (unsigned)i; rank = rank - cum; break; }
        cum += c;
    }
    unsigned prefix = (pass == 0) ? b : ((sel[0] << 10) | b);
    sel[0] = prefix;
    sel[1] = (unsigned)rank;
    if (pass == 2) ((float*)sel)[2] = unkey(prefix);   // exact kk-th largest value
    for (int i = 0; i < nb; i++) hist[i] = 0u;         // keep graph-replay deterministic
}

// ---------------- Wh = h @ W via V_WMMA_F32_16X16X4_F32 ----------------
__global__ void wh_gemm_kernel(const float* __restrict__ h, const float* __restrict__ W,
                               float* __restrict__ Wh) {
    int lane = threadIdx.x;
    int l = lane & 15, g = lane >> 4;
    int m0 = blockIdx.x * 16;
    int n0 = blockIdx.y * 16;
    v8f acc = {0.f, 0.f, 0.f, 0.f, 0.f, 0.f, 0.f, 0.f};
    const float* hrow = h + (long long)(m0 + l) * IN_FEAT;
    for (int k0 = 0; k0 < IN_FEAT; k0 += 4) {
        v2f a, b;
        a.x = hrow[k0 + 2 * g];
        a.y = hrow[k0 + 2 * g + 1];
        b.x = W[(k0 + 2 * g) * OUT_FEAT + n0 + l];
        b.y = W[(k0 + 2 * g + 1) * OUT_FEAT + n0 + l];
        acc = __builtin_amdgcn_wmma_f32_16x16x4_f32(false, a, false, b, (short)0, acc,
                                                    false, false);
    }
#pragma unroll
    for (int r = 0; r < 8; r++)
        Wh[(long long)(m0 + r + 8 * g) * OUT_FEAT + n0 + l] = acc[r];
}

// ---------------- s1/s2 = Wh @ a[:F], Wh @ a[F:] ----------------
__global__ void svec_kernel(const float* __restrict__ Wh, const float* __restrict__ a,
                            float* __restrict__ s1, float* __restrict__ s2) {
    int row = blockIdx.x * blockDim.x + threadIdx.x;
    if (row >= N_NODES) return;
    float a1 = 0.f, a2 = 0.f;
    const float* w = Wh + (long long)row * OUT_FEAT;
    for (int f = 0; f < OUT_FEAT; f++) {
        float v = w[f];
        a1 += v * a[f];
        a2 += v * a[OUT_FEAT + f];
    }
    s1[row] = a1;
    s2[row] = a2;
}

// ---------------- fused mask + online softmax + attention@Wh ----------------
__global__ void attention_kernel(const float* __restrict__ adj, const float* __restrict__ Wh,
                                 const float* __restrict__ s1, const float* __restrict__ s2,
                                 const float* __restrict__ thrp, float* __restrict__ out_h,
                                 float* __restrict__ out_adj) {
    __shared__ float Pt[2][16 * PSTR];
    __shared__ float red[16][17];
    __shared__ float rowmax[16], rowsum[16], rowscale[16], s1s[16];

    int t = threadIdx.x;
    int i0 = blockIdx.x * 16;
    float thr = *thrp;

    int lane = t & 31;
    int wave = t >> 5;
    int l = lane & 15, g = lane >> 4;
    int n0 = wave * 16;
    int tx = t & 127, ty = t >> 7;       // transform mapping: 2 thread-rows x 128 cols
    int rr = t >> 4, seg = t & 15;       // reduction mapping: 16 rows x 16 segments

    if (t < 16) { s1s[t] = s1[i0 + t]; rowmax[t] = -3e38f; rowsum[t] = 0.f; }
    v8f acc = {0.f, 0.f, 0.f, 0.f, 0.f, 0.f, 0.f, 0.f};

    // prefill chunk 0 into buffer 0: 512 x 16B async transfers (2 per thread)
#pragma unroll
    for (int i = t; i < 512; i += 256) {
        int row = i >> 5, sg = i & 31;
        const float* gaddr = adj + (long long)(i0 + row) * N_NODES + sg * 4;
        async_g2l_b128(lds_addr32(&Pt[0][row * PSTR + sg * 4]), gaddr);
    }

    int c = 0;
    for (int j0 = 0; j0 < N_NODES; j0 += CHUNK, ++c) {
        int buf = c & 1;
        bool hasnext = (j0 + CHUNK < N_NODES);
        if (hasnext) {
#pragma unroll
            for (int i = t; i < 512; i += 256) {
                int row = i >> 5, sg = i & 31;
                const float* gaddr =
                    adj + (long long)(i0 + row) * N_NODES + (j0 + CHUNK) + sg * 4;
                async_g2l_b128(lds_addr32(&Pt[buf ^ 1][row * PSTR + sg * 4]), gaddr);
            }
            asm volatile("s_wait_asynccnt 0x2" ::: "memory");  // current chunk landed
        } else {
            asm volatile("s_wait_asynccnt 0x0" ::: "memory");
        }
        __syncthreads();  // all waves' async fills of Pt[buf] visible

        // transform in place: raw adj -> masked score; emit adj_reconstructed
        float s2v = s2[j0 + tx];
        for (int r = ty; r < 16; r += 2) {
            float av = Pt[buf][r * PSTR + tx];
            float e = leaky(s1s[r] + s2v);
            float m = (av >= thr) ? e : NEG_FILL;
            Pt[buf][r * PSTR + tx] = m;
            out_adj[(long long)(i0 + r) * N_NODES + j0 + tx] = m;
        }
        __syncthreads();

        // chunk max per row (two-level reduction)
        float mx = -3e38f;
#pragma unroll
        for (int cc = 0; cc < 8; cc++) mx = fmaxf(mx, Pt[buf][rr * PSTR + seg * 8 + cc]);
        red[rr][seg] = mx;
        __syncthreads();
        if (t < 16) {
            float m_old = rowmax[t];
            float cm = -3e38f;
            for (int s = 0; s < 16; s++) cm = fmaxf(cm, red[t][s]);
            float m_new = fmaxf(m_old, cm);
            float sc = __expf(m_old - m_new);
            rowscale[t] = sc;
            rowsum[t] *= sc;
            rowmax[t] = m_new;
        }
        __syncthreads();

        // exponentiate in place, accumulate row sums
        float mrow = rowmax[rr];
        float ls = 0.f;
#pragma unroll
        for (int cc = 0; cc < 8; cc++) {
            float p = __expf(Pt[buf][rr * PSTR + seg * 8 + cc] - mrow);
            Pt[buf][rr * PSTR + seg * 8 + cc] = p;
            ls += p;
        }
        red[rr][seg] = ls;
        __syncthreads();
        if (t < 16) {
            float s = 0.f;
            for (int si = 0; si < 16; si++) s += red[t][si];
            rowsum[t] += s;
        }

        // flash rescale of accumulators, then P_chunk @ Wh_chunk via WMMA
#pragma unroll
        for (int r = 0; r < 8; r++) acc[r] *= rowscale[r + 8 * g];
        for (int kk = 0; kk < CHUNK; kk += 4) {
            v2f a, b;
            a.x = Pt[buf][l * PSTR + kk + 2 * g];
            a.y = Pt[buf][l * PSTR + kk + 2 * g + 1];
            b.x = Wh[(long long)(j0 + kk + 2 * g) * OUT_FEAT + n0 + l];
            b.y = Wh[(long long)(j0 + kk + 2 * g + 1) * OUT_FEAT + n0 + l];
            acc = __builtin_amdgcn_wmma_f32_16x16x4_f32(false, a, false, b, (short)0, acc,
                                                        false, false);
        }
        __syncthreads();  // WMMA ds reads done before buf is refilled next iteration
    }

    // finalize: normalize and apply output LeakyReLU
#pragma unroll
    for (int r = 0; r < 8; r++) {
        int row = r + 8 * g;
        float v = acc[r] / rowsum[row];
        out_h[(long long)(i0 + row) * OUT_FEAT + n0 + l] = leaky(v);
    }
}

extern "C" void kernel_launch(void* const* d_in, const int* in_sizes, int n_in,
                              void* d_out, int out_size, void* d_ws, size_t ws_size,
                              hipStream_t stream) {
    const float* h   = (const float*)d_in[0];
    const float* adj = (const float*)d_in[1];
    const float* W   = (const float*)d_in[2];
    const float* a   = (const float*)d_in[3];
    const int*   kp  = (const int*)d_in[4];

    float* out     = (float*)d_out;
    float* out_h   = out;                                  // [8192,128]
    float* out_adj = out + (long long)N_NODES * OUT_FEAT;  // [8192,8192]

    float*    Wh   = (float*)d_ws;                       // 8192*128
    float*    s1   = Wh + (long long)N_NODES * OUT_FEAT; // 8192
    float*    s2   = s1 + N_NODES;                       // 8192
    unsigned* hist = (unsigned*)(s2 + N_NODES);          // 4096 bins
    unsigned* sel  = hist + 4096;                        // prefix, rank, thr(float)

    zero_kernel<<<16, 256, 0, stream>>>(hist, 4096 + 8);

    hist_kernel<<<1024, 256, 0, stream>>>(adj, hist, sel, 0);
    scan_kernel<<<1, 32, 0, stream>>>(hist, sel, kp, 0);
    hist_kernel<<<1024, 256, 0, stream>>>(adj, hist, sel, 1);
    scan_kernel<<<1, 32, 0, stream>>>(hist, sel, kp, 1);
    hist_kernel<<<1024, 256, 0, stream>>>(adj, hist, sel, 2);
    scan_kernel<<<1, 32, 0, stream>>>(hist, sel, kp, 2);

    dim3 g1(N_NODES / 16, OUT_FEAT / 16);
    wh_gemm_kernel<<<g1, 32, 0, stream>>>(h, W, Wh);
    svec_kernel<<<N_NODES / 256, 256, 0, stream>>>(Wh, a, s1, s2);

    const float* thrp = (const float*)(sel + 2);
    attention_kernel<<<N_NODES / 16, 256, 0, stream>>>(adj, Wh, s1, s2, thrp, out_h, out_adj);
}